// GATLayer_21303037788170
// MI455X (gfx1250) — compile-verified
//
#include <hip/hip_runtime.h>
#include <hip/hip_bf16.h>

#define IN_DIM  128
#define OUT_DIM 64

typedef __attribute__((ext_vector_type(2))) float v2f;
typedef __attribute__((ext_vector_type(8))) float v8f;

// ---------------------------------------------------------------------------
// Kernel 0: init max-bits to -inf, denom to 0, out to 0
// ---------------------------------------------------------------------------
__global__ void gat_init_kernel(unsigned int* __restrict__ maxbits,
                                float* __restrict__ denom,
                                float* __restrict__ out,
                                int n_nodes) {
    int i = blockIdx.x * blockDim.x + threadIdx.x;
    int total = n_nodes * OUT_DIM;
    if (i < total) out[i] = 0.0f;
    if (i < n_nodes) {
        maxbits[i] = 0xFF800000u;   // bits of -inf
        denom[i]   = 0.0f;
    }
}

// ---------------------------------------------------------------------------
// Kernel 1: z = h @ W^T via V_WMMA_F32_16X16X4_F32
//   block = 128 threads = 4 waves; wave w computes the 16x16 tile
//   (rows row0..row0+15) x (cols 16w..16w+15). K=128 in 32 unrolled steps.
// A layout (16x4 f32): lanes 0-15 = M rows, VGPR0/1 = {K, K+1}; lanes 16-31
// hold {K+2, K+3}.  B layout (4x16): lanes = N cols, same K split.
// ---------------------------------------------------------------------------
__global__ void __launch_bounds__(128)
gat_gemm_wmma_kernel(const float* __restrict__ h,
                     const float* __restrict__ W,
                     float* __restrict__ z,
                     int n_nodes) {
    const int lane  = threadIdx.x & 31;
    const int wave  = threadIdx.x >> 5;          // 0..3  -> n-tile
    const int half  = lane >> 4;                 // 0 or 1 (K split)
    const int lcol  = lane & 15;                 // M row (A) / N col (B)

    const int row0  = blockIdx.x * 16;
    const int nbase = wave * 16;

    int arow = row0 + lcol;
    if (arow >= n_nodes) arow = n_nodes - 1;     // clamp, keep EXEC all-ones
    const int ncol = nbase + lcol;

    const float* __restrict__ hrow = h + (size_t)arow * IN_DIM;
    const float* __restrict__ wrow = W + (size_t)ncol * IN_DIM;

    v8f acc = {};
#pragma unroll
    for (int k = 0; k < IN_DIM; k += 4) {
        const int kk = k + 2 * half;             // even -> 8B aligned
        float2 af = *(const float2*)(hrow + kk);
        float2 bf = *(const float2*)(wrow + kk);
        v2f a; a[0] = af.x; a[1] = af.y;
        v2f b; b[0] = bf.x; b[1] = bf.y;
        acc = __builtin_amdgcn_wmma_f32_16x16x4_f32(
            /*neg_a=*/false, a, /*neg_b=*/false, b,
            /*c_mod=*/(short)0, acc, /*reuse_a=*/false, /*reuse_b=*/false);
    }

    // C/D layout: VGPR v, lanes 0-15 -> M=v, lanes 16-31 -> M=v+8; N=lane&15
#pragma unroll
    for (int v = 0; v < 8; ++v) {
        int row = row0 + v + 8 * half;
        if (row < n_nodes)
            z[(size_t)row * OUT_DIM + ncol] = acc[v];
    }
}

// ---------------------------------------------------------------------------
// Kernel 2: per-edge score e = <z[src], z[dst]>, one wave per edge.
//   Lane l loads dims {2l, 2l+1} (contiguous 256B per row -> coalesced, L2
//   resident: z is 25.6MB << 192MB L2). Wave-reduce, lane 0 writes score and
//   does float atomic-max via signed/unsigned int ordering trick.
// ---------------------------------------------------------------------------
__global__ void __launch_bounds__(256)
gat_edge_score_kernel(const float* __restrict__ z,
                      const int* __restrict__ src,
                      const int* __restrict__ dst,
                      float* __restrict__ e,
                      unsigned int* __restrict__ maxbits,
                      int n_edges) {
    const int lane = threadIdx.x & 31;
    const int eid  = blockIdx.x * (blockDim.x >> 5) + (threadIdx.x >> 5);
    if (eid >= n_edges) return;

    const int s = src[eid];
    const int d = dst[eid];
    const float2* zs = (const float2*)(z + (size_t)s * OUT_DIM);
    const float2* zd = (const float2*)(z + (size_t)d * OUT_DIM);
    float2 a = zs[lane];
    float2 b = zd[lane];
    float acc = a.x * b.x + a.y * b.y;
#pragma unroll
    for (int off = 16; off > 0; off >>= 1)
        acc += __shfl_xor(acc, off, 32);

    if (lane == 0) {
        e[eid] = acc;
        // monotone float max: raw bits, signed cmp for >=0, unsigned for <0
        if (acc >= 0.0f)
            atomicMax((int*)&maxbits[d], __float_as_int(acc));
        else
            atomicMin(&maxbits[d], __float_as_uint(acc));
    }
}

// ---------------------------------------------------------------------------
// Kernel 3: e_exp = exp(e - max[dst]) in place; denom[dst] += e_exp
// ---------------------------------------------------------------------------
__global__ void gat_exp_denom_kernel(const int* __restrict__ dst,
                                     float* __restrict__ e,
                                     const unsigned int* __restrict__ maxbits,
                                     float* __restrict__ denom,
                                     int n_edges) {
    int eid = blockIdx.x * blockDim.x + threadIdx.x;
    if (eid >= n_edges) return;
    int d = dst[eid];
    float m = __int_as_float((int)maxbits[d]);   // finite: segment non-empty
    float t = __expf(e[eid] - m);
    e[eid] = t;
    atomicAdd(&denom[d], t);
}

// ---------------------------------------------------------------------------
// Kernel 4: out[dst] += (e_exp / denom[dst]) * z[src]
//   64 lanes per edge (dims 0..63): contiguous 256B gather of z[src] and a
//   contiguous 256B burst of f32 atomic adds into out[dst] (L2-resident).
// ---------------------------------------------------------------------------
__global__ void __launch_bounds__(256)
gat_aggregate_kernel(const float* __restrict__ z,
                     const int* __restrict__ src,
                     const int* __restrict__ dst,
                     const float* __restrict__ e,
                     const float* __restrict__ denom,
                     float* __restrict__ out,
                     int n_edges) {
    const int j   = threadIdx.x & 63;
    const int eid = blockIdx.x * (blockDim.x >> 6) + (threadIdx.x >> 6);
    if (eid >= n_edges) return;
    const int s = src[eid];
    const int d = dst[eid];
    const float alpha = e[eid] / fmaxf(denom[d], 1e-30f);
    atomicAdd(&out[(size_t)d * OUT_DIM + j], alpha * z[(size_t)s * OUT_DIM + j]);
}

// ---------------------------------------------------------------------------
extern "C" void kernel_launch(void* const* d_in, const int* in_sizes, int n_in,
                              void* d_out, int out_size, void* d_ws, size_t ws_size,
                              hipStream_t stream) {
    const float* h   = (const float*)d_in[0];
    const float* W   = (const float*)d_in[1];
    const int*   src = (const int*)d_in[2];
    const int*   dst = (const int*)d_in[3];
    float* out = (float*)d_out;

    const int n_nodes = in_sizes[0] / IN_DIM;    // 100000
    const int n_edges = in_sizes[2];             // 1600000

    // workspace layout
    float*        z       = (float*)d_ws;                        // n_nodes*64
    float*        e       = z + (size_t)n_nodes * OUT_DIM;       // n_edges
    unsigned int* maxbits = (unsigned int*)(e + n_edges);        // n_nodes
    float*        denom   = (float*)(maxbits + n_nodes);         // n_nodes

    // 0) init
    {
        int total = n_nodes * OUT_DIM;
        gat_init_kernel<<<(total + 255) / 256, 256, 0, stream>>>(
            maxbits, denom, out, n_nodes);
    }
    // 1) z = h @ W^T (WMMA f32 16x16x4)
    {
        int blocks = (n_nodes + 15) / 16;
        gat_gemm_wmma_kernel<<<blocks, 128, 0, stream>>>(h, W, z, n_nodes);
    }
    // 2) edge scores + segment max (wave per edge)
    {
        int edges_per_block = 256 / 32;
        int blocks = (n_edges + edges_per_block - 1) / edges_per_block;
        gat_edge_score_kernel<<<blocks, 256, 0, stream>>>(
            z, src, dst, e, maxbits, n_edges);
    }
    // 3) exp + segment sum
    {
        gat_exp_denom_kernel<<<(n_edges + 255) / 256, 256, 0, stream>>>(
            dst, e, maxbits, denom, n_edges);
    }
    // 4) weighted aggregation scatter (64 lanes per edge)
    {
        int edges_per_block = 256 / 64;
        int blocks = (n_edges + edges_per_block - 1) / edges_per_block;
        gat_aggregate_kernel<<<blocks, 256, 0, stream>>>(
            z, src, dst, e, denom, out, n_edges);
    }
}